// SPPostProcessingBlock_53266184405090
// MI455X (gfx1250) — compile-verified
//
#include <hip/hip_runtime.h>

// out[i] = adj[i] > 0.5f ? 1.0f : 0.0f   (closed form of the 1000-iter prox loop)
//
// Streaming pipeline (CDNA5): double-buffered GLOBAL_LOAD_ASYNC_TO_LDS_B128
// (ASYNCcnt) -> ds_load_b128 -> branchless select -> nontemporal b128 store.

typedef float v4f __attribute__((ext_vector_type(4)));

#define BLOCK 256
#define THRESH 0.5f

__global__ __launch_bounds__(BLOCK) void sp_thresh_async(
    const v4f* __restrict__ in4,
    v4f* __restrict__ out4,
    const float* __restrict__ in,
    float* __restrict__ out,
    unsigned n, unsigned n4, unsigned ntiles)
{
    __shared__ v4f tile[2][BLOCK];

    const unsigned tid    = threadIdx.x;
    const unsigned stride = gridDim.x;

    // Raw LDS byte offsets of this lane's two buffer slots.
    // On AMDGPU, generic->LDS addrspacecast is truncation to the low 32 bits,
    // which is exactly the wave-relative LDS byte address the async op wants.
    unsigned lds_slot[2];
    lds_slot[0] = (unsigned)(unsigned long long)&tile[0][tid];
    lds_slot[1] = (unsigned)(unsigned long long)&tile[1][tid];

    const unsigned t0 = blockIdx.x;
    unsigned par = 0;

    // Prologue: issue async load of this block's first tile into buffer 0.
    // OOB lanes clamp their address (issue stays wave-uniform so ASYNCcnt
    // accounting is exact: one async instruction per wave per tile).
    if (t0 < ntiles) {
        unsigned g = t0 * BLOCK + tid;
        if (g >= n4) g = n4 - 1u;
        const v4f* p = in4 + g;
        asm volatile("global_load_async_to_lds_b128 %0, %1, off"
                     :: "v"(lds_slot[0]), "v"(p) : "memory");
    }

    for (unsigned k = t0; k < ntiles; k += stride) {
        const unsigned nk = k + stride;
        if (nk < ntiles) {
            // Issue next tile into the other buffer, then wait until only
            // that newest async load is still outstanding (current tile done).
            unsigned g = nk * BLOCK + tid;
            if (g >= n4) g = n4 - 1u;
            const v4f* p = in4 + g;
            asm volatile("global_load_async_to_lds_b128 %0, %1, off"
                         :: "v"(lds_slot[par ^ 1u]), "v"(p) : "memory");
            asm volatile("s_wait_asynccnt 0x1" ::: "memory");
        } else {
            asm volatile("s_wait_asynccnt 0x0" ::: "memory");
        }

        const unsigned g = k * BLOCK + tid;
        if (g < n4) {
            v4f v = tile[par][tid];          // ds_load_b128
            v4f r;
            r.x = (v.x > THRESH) ? 1.0f : 0.0f;
            r.y = (v.y > THRESH) ? 1.0f : 0.0f;
            r.z = (v.z > THRESH) ? 1.0f : 0.0f;
            r.w = (v.w > THRESH) ? 1.0f : 0.0f;
            __builtin_nontemporal_store(r, out4 + g);  // streamed once, NT store
        }
        par ^= 1u;
    }

    // Scalar tail (n % 4 != 0) — block 0 mops up. (Empty for 2*1024*1024.)
    if (blockIdx.x == 0) {
        for (unsigned i = n4 * 4u + tid; i < n; i += BLOCK) {
            float x = in[i];
            out[i] = (x > THRESH) ? 1.0f : 0.0f;
        }
    }
}

extern "C" void kernel_launch(void* const* d_in, const int* in_sizes, int n_in,
                              void* d_out, int out_size, void* d_ws, size_t ws_size,
                              hipStream_t stream)
{
    const float* in  = (const float*)d_in[0];
    float*       out = (float*)d_out;

    const unsigned n      = (unsigned)in_sizes[0];      // 2*1024*1024
    const unsigned n4     = n / 4u;
    const unsigned ntiles = (n4 + BLOCK - 1u) / BLOCK;  // 2048 for this shape

    unsigned grid = 512u;                  // 4 tiles/block -> real double-buffering
    if (ntiles < grid) grid = ntiles ? ntiles : 1u;

    sp_thresh_async<<<dim3(grid), dim3(BLOCK), 0, stream>>>(
        (const v4f*)in, (v4f*)out, in, out, n, n4, ntiles);
}